// CompactControlAttention_64244120814152
// MI455X (gfx1250) — compile-verified
//
#include <hip/hip_runtime.h>

// ---------------------------------------------------------------------------
// CompactControlAttention, collapsed form (see analysis):
//   softmax over size-1 axis == 1.0  =>  Wq/bq/Wk/bk are dead.
//   out[b,s,:] = seq[b,s,:] + ( (sum_c controls[c,b,:]) @ Wv^T + C*bv ) @ Wo^T + C*bo
// ---------------------------------------------------------------------------

typedef float v2f __attribute__((ext_vector_type(2)));
typedef float v4f __attribute__((ext_vector_type(4)));
typedef float v8f __attribute__((ext_vector_type(8)));

constexpr int Dm = 2048;  // d_model
constexpr int Bm = 64;    // batch
constexpr int Sm = 32;    // seq_len
constexpr int Cm = 8;     // controls

// ---- Step 1: cs[b,d] = sum_c controls[c,b,d]  (float4 coalesced) -----------
__global__ __launch_bounds__(256) void cc_reduce_controls(
    const float* __restrict__ ctrl, float* __restrict__ cs) {
  const int BD4 = Bm * Dm / 4;
  int i = blockIdx.x * blockDim.x + threadIdx.x;
  if (i >= BD4) return;
  const v4f* c4 = (const v4f*)ctrl;
  v4f s = c4[i];
#pragma unroll
  for (int c = 1; c < Cm; ++c) s += c4[(size_t)c * BD4 + i];
  ((v4f*)cs)[i] = s;
}

// ---- Step 2/3: out[m,n] = sum_k A[m,k]*W[n,k] + biasScale*bias[n] ----------
// One wave32 per 16x16 tile, fp32 WMMA (V_WMMA_F32_16X16X4_F32), K swept by 4.
// ISA 32-bit A 16x4 layout: lane t holds A[t&15][k + 2*(t>>4) + {0,1}] in 2 VGPRs.
// B (4x16, K-major) mirrors it with row = output column n (since B[k][n]=W[n][k]).
// C/D layout: VGPR i -> row (t>>4)*8 + i, col t&15.
__global__ __launch_bounds__(256) void cc_gemm_tn_wmma(
    const float* __restrict__ A, const float* __restrict__ W,
    const float* __restrict__ bias, float* __restrict__ out,
    int M, int N, int K, float biasScale) {
  const int lane = threadIdx.x & 31;
  const int wave = threadIdx.x >> 5;
  const int nTiles = N >> 4;
  const int tile = blockIdx.x * 8 + wave;      // 8 waves per 256-thread block
  if (tile >= (M >> 4) * nTiles) return;       // wave-uniform guard (EXEC stays all-1s)
  const int mTile = tile / nTiles;             // waves in a block share mTile -> A reuse
  const int nTile = tile % nTiles;

  const int r    = lane & 15;                  // A row within tile / B column within tile
  const int kSub = (lane >> 4) << 1;           // 0 or 2: K sub-offset per lane half
  const float* aRow = A + (size_t)(mTile * 16 + r) * K + kSub;
  const float* wRow = W + (size_t)(nTile * 16 + r) * K + kSub;

  v8f acc = {0.f, 0.f, 0.f, 0.f, 0.f, 0.f, 0.f, 0.f};
  for (int k0 = 0; k0 < K; k0 += 32) {
    __builtin_prefetch(wRow + k0 + 128, 0, 0);  // global_prefetch_b8 on streamed weights
    __builtin_prefetch(aRow + k0 + 128, 0, 0);
#pragma unroll
    for (int kk = 0; kk < 32; kk += 4) {
      v2f a = *(const v2f*)(aRow + k0 + kk);   // 8B load: A[m][k..k+1] (per lane half)
      v2f b = *(const v2f*)(wRow + k0 + kk);   // 8B load: W[n][k..k+1]
      acc = __builtin_amdgcn_wmma_f32_16x16x4_f32(
          /*neg_a=*/false, a, /*neg_b=*/false, b,
          /*c_mod=*/(short)0, acc, /*reuse_a=*/false, /*reuse_b=*/false);
    }
  }

  const int nCol  = nTile * 16 + r;
  const float badd = bias[nCol] * biasScale;
  const int mBase = mTile * 16 + ((lane >> 4) << 3);
#pragma unroll
  for (int i = 0; i < 8; ++i)
    out[(size_t)(mBase + i) * N + nCol] = acc[i] + badd;
}

// ---- Step 4: out[b,s,:] = seq[b,s,:] + P[b,:]  (float4 coalesced) ----------
__global__ __launch_bounds__(256) void cc_residual(
    const float* __restrict__ seq, const float* __restrict__ P,
    float* __restrict__ out) {
  const int d4 = Dm / 4;
  const int tot = Bm * Sm * d4;
  int i = blockIdx.x * blockDim.x + threadIdx.x;   // float4 index over (B,S,D/4)
  if (i >= tot) return;
  int b  = i / (Sm * d4);
  int e4 = i % d4;
  v4f r = ((const v4f*)seq)[i] + ((const v4f*)P)[(size_t)b * d4 + e4];
  ((v4f*)out)[i] = r;
}

// ---------------------------------------------------------------------------
extern "C" void kernel_launch(void* const* d_in, const int* in_sizes, int n_in,
                              void* d_out, int out_size, void* d_ws, size_t ws_size,
                              hipStream_t stream) {
  // setup_inputs order: sequence, controls, Wq, bq, Wk, bk, Wv, bv, Wo, bo
  const float* seq  = (const float*)d_in[0];
  const float* ctrl = (const float*)d_in[1];
  // d_in[2..5] (Wq,bq,Wk,bk) are mathematically dead (softmax over size-1 axis == 1)
  const float* Wv   = (const float*)d_in[6];
  const float* bv   = (const float*)d_in[7];
  const float* Wo   = (const float*)d_in[8];
  const float* bo   = (const float*)d_in[9];
  float* out = (float*)d_out;

  float* cs = (float*)d_ws;               // 64*2048 fp32 = 512 KB
  float* U  = cs + (size_t)Bm * Dm;       // 512 KB
  float* P  = U  + (size_t)Bm * Dm;       // 512 KB   (total 1.5 MB of d_ws)

  {  // 1) sum controls over C
    int n4 = Bm * Dm / 4;
    cc_reduce_controls<<<(n4 + 255) / 256, 256, 0, stream>>>(ctrl, cs);
  }
  {  // 2) U = cs @ Wv^T + C*bv ; 3) P = U @ Wo^T + C*bo
    int waves  = (Bm / 16) * (Dm / 16);   // 4 * 128 = 512 tiles
    int blocks = (waves + 7) / 8;         // 8 waves per block
    cc_gemm_tn_wmma<<<blocks, 256, 0, stream>>>(cs, Wv, bv, U, Bm, Dm, Dm, (float)Cm);
    cc_gemm_tn_wmma<<<blocks, 256, 0, stream>>>(U,  Wo, bo, P, Bm, Dm, Dm, (float)Cm);
  }
  {  // 4) broadcast residual add over S
    int n4 = Bm * Sm * Dm / 4;
    cc_residual<<<(n4 + 255) / 256, 256, 0, stream>>>(seq, P, out);
  }
}